// Rank_loss_F_41678362640810
// MI455X (gfx1250) — compile-verified
//
#include <hip/hip_runtime.h>
#include <hip/hip_bf16.h>

// ---------------------------------------------------------------------------
// Rank loss: normalize -> pairwise dist via fp32 WMMA -> masked row reductions
// N=8192, D=512. dist written to d_out+1, scalar loss to d_out[0].
// ---------------------------------------------------------------------------

#define NROWS 8192
#define DIMS  512
#define EPSV  1e-5f

// GEMM tiling
#define BM 128
#define BN 64
#define BK 32
#define LDA 36   // LDS row stride (dwords): 16B-aligned float4 stores, conflict-free frag reads

typedef __attribute__((ext_vector_type(2))) float v2f;
typedef __attribute__((ext_vector_type(8))) float v8f;

// ---------------------------------------------------------------------------
// Kernel 1: L2 normalize each row, record sq[i] = sum(xn^2)
// ---------------------------------------------------------------------------
__global__ __launch_bounds__(256) void normalize_kernel(
    const float* __restrict__ x, float* __restrict__ xn, float* __restrict__ sq)
{
    const int row  = blockIdx.x;
    const int tid  = threadIdx.x;
    const int lane = tid & 31;
    const int wave = tid >> 5;

    const float* xr = x + (size_t)row * DIMS;
    float v0 = xr[tid];
    float v1 = xr[tid + 256];
    float ss = v0 * v0 + v1 * v1;

    #pragma unroll
    for (int off = 16; off; off >>= 1) ss += __shfl_down(ss, off, 32);

    __shared__ float red[8];
    __shared__ float rinv_sh;
    if (lane == 0) red[wave] = ss;
    __syncthreads();
    if (tid == 0) {
        float t = 0.f;
        #pragma unroll
        for (int w = 0; w < 8; ++w) t += red[w];
        float nrm = sqrtf(t);
        float cn  = fmaxf(nrm, 1e-12f);
        float ri  = 1.0f / cn;
        rinv_sh   = ri;
        sq[row]   = t * ri * ri;
    }
    __syncthreads();
    float ri = rinv_sh;
    float* xo = xn + (size_t)row * DIMS;
    xo[tid]       = v0 * ri;
    xo[tid + 256] = v1 * ri;
}

// ---------------------------------------------------------------------------
// Kernel 2: dist tile via V_WMMA_F32_16X16X4_F32.
// Block tile 128x64, 8 waves (4x2), wave tile 32x32 = 2x2 WMMA accumulators.
// ---------------------------------------------------------------------------
__global__ __launch_bounds__(256) void dist_gemm_kernel(
    const float* __restrict__ xn, const float* __restrict__ sq,
    float* __restrict__ dist)
{
    __shared__ float As[BM * LDA];
    __shared__ float Bs[BN * LDA];
    __shared__ float sqR[BM];
    __shared__ float sqC[BN];

    const int tid   = threadIdx.x;
    const int lane  = tid & 31;
    const int wave  = tid >> 5;
    const int waveR = wave >> 1;   // 0..3
    const int waveC = wave & 1;    // 0..1
    const int rowBase = blockIdx.y * BM;
    const int colBase = blockIdx.x * BN;

    if (tid < BM) sqR[tid] = sq[rowBase + tid];
    if (tid < BN) sqC[tid] = sq[colBase + tid];

    v8f acc[2][2] = {};

    const int half = lane >> 4;     // 0: K={k,k+1}, 1: K={k+2,k+3}
    const int l15  = lane & 15;
    const int ldr  = tid >> 3;          // 0..31
    const int ldc  = (tid & 7) * 4;     // 0,4,...,28

    for (int k0 = 0; k0 < DIMS; k0 += BK) {
        __syncthreads();
        // Stage A (128x32) and B (64x32) into LDS, coalesced float4.
        #pragma unroll
        for (int p = 0; p < 4; ++p) {
            float4 v = *(const float4*)&xn[(size_t)(rowBase + ldr + 32 * p) * DIMS + k0 + ldc];
            *(float4*)&As[(ldr + 32 * p) * LDA + ldc] = v;
        }
        #pragma unroll
        for (int p = 0; p < 2; ++p) {
            float4 v = *(const float4*)&xn[(size_t)(colBase + ldr + 32 * p) * DIMS + k0 + ldc];
            *(float4*)&Bs[(ldr + 32 * p) * LDA + ldc] = v;
        }
        __syncthreads();

        #pragma unroll
        for (int kk = 0; kk < BK; kk += 4) {
            v2f aF[2], bF[2];
            #pragma unroll
            for (int tr = 0; tr < 2; ++tr)
                aF[tr] = *(const v2f*)&As[(waveR * 32 + tr * 16 + l15) * LDA + kk + 2 * half];
            #pragma unroll
            for (int tc = 0; tc < 2; ++tc)
                bF[tc] = *(const v2f*)&Bs[(waveC * 32 + tc * 16 + l15) * LDA + kk + 2 * half];
            #pragma unroll
            for (int tr = 0; tr < 2; ++tr)
                #pragma unroll
                for (int tc = 0; tc < 2; ++tc)
                    acc[tr][tc] = __builtin_amdgcn_wmma_f32_16x16x4_f32(
                        false, aF[tr], false, bF[tc],
                        (short)0, acc[tr][tc], false, false);
        }
    }

    // Epilogue: Gram -> Euclidean distance, per-lane b32 stores per C layout.
    #pragma unroll
    for (int tr = 0; tr < 2; ++tr) {
        #pragma unroll
        for (int tc = 0; tc < 2; ++tc) {
            #pragma unroll
            for (int v = 0; v < 8; ++v) {
                int mi = waveR * 32 + tr * 16 + v + 8 * half;
                int nj = waveC * 32 + tc * 16 + l15;
                float g  = acc[tr][tc][v];
                float d2 = sqR[mi] + sqC[nj] - 2.0f * g;
                float d  = sqrtf(fmaxf(d2, 1e-12f));
                dist[(size_t)(rowBase + mi) * NROWS + (colBase + nj)] = d;
            }
        }
    }
}

// ---------------------------------------------------------------------------
// Kernel 3: per-row masked reductions -> per-row loss (deterministic).
// ---------------------------------------------------------------------------
__global__ __launch_bounds__(256) void row_reduce_kernel(
    const float* __restrict__ dist, const int* __restrict__ targets,
    const int* __restrict__ sub, float* __restrict__ rowLoss)
{
    const int i    = blockIdx.x;
    const int tid  = threadIdx.x;
    const int lane = tid & 31;
    const int wave = tid >> 5;

    const int ti = targets[i];
    const int si = sub[i];

    float ap1 = 0.f, c1 = 0.f, ap2 = 0.f, c2 = 0.f;
    float an1n = 0.f, an1d = 0.f, an2n = 0.f, an2d = 0.f;

    for (int j = tid; j < NROWS; j += 256) {
        float d = dist[(size_t)i * NROWS + j];
        int tj = targets[j];
        int sj = sub[j];
        bool pos   = (ti == tj) && (j != i);   // diagonal excluded from positives
        bool intra = (si == sj);
        if (pos) {
            if (intra) { ap1 += fmaxf(d - 1.7f, 0.f); c1 += 1.f; }   // margin1-alpha1 = -1.7
            else       { ap2 += fmaxf(d - 0.7f, 0.f); c2 += 1.f; }   // margin2-alpha2 = -0.7
        } else if (d < 2.2f) {                  // negatives include the diagonal
            float a = 2.2f - d;
            float w = __expf(a);                // TVAL = 1.0
            float t = a * w;
            if (intra) { an1n += t; an1d += w; }
            else       { an2n += t; an2d += w; }
        }
    }

    float vals[8] = { ap1, c1, ap2, c2, an1n, an1d, an2n, an2d };
    __shared__ float red[8][8];
    #pragma unroll
    for (int k = 0; k < 8; ++k) {
        float v = vals[k];
        #pragma unroll
        for (int off = 16; off; off >>= 1) v += __shfl_down(v, off, 32);
        if (lane == 0) red[wave][k] = v;
    }
    __syncthreads();
    if (tid == 0) {
        float s[8] = {0.f, 0.f, 0.f, 0.f, 0.f, 0.f, 0.f, 0.f};
        for (int w = 0; w < 8; ++w)
            #pragma unroll
            for (int k = 0; k < 8; ++k) s[k] += red[w][k];
        rowLoss[i] = s[0] / (s[1] + EPSV) + s[2] / (s[3] + EPSV)
                   + s[4] / (s[5] + EPSV) + s[6] / (s[7] + EPSV);
    }
}

// ---------------------------------------------------------------------------
// Kernel 4: mean over rows -> d_out[0]
// ---------------------------------------------------------------------------
__global__ __launch_bounds__(256) void final_loss_kernel(
    const float* __restrict__ rowLoss, float* __restrict__ out)
{
    const int tid  = threadIdx.x;
    const int lane = tid & 31;
    const int wave = tid >> 5;
    float s = 0.f;
    for (int i = tid; i < NROWS; i += 256) s += rowLoss[i];
    #pragma unroll
    for (int off = 16; off; off >>= 1) s += __shfl_down(s, off, 32);
    __shared__ float red[8];
    if (lane == 0) red[wave] = s;
    __syncthreads();
    if (tid == 0) {
        float t = 0.f;
        #pragma unroll
        for (int w = 0; w < 8; ++w) t += red[w];
        out[0] = t / (float)NROWS;
    }
}

// ---------------------------------------------------------------------------
extern "C" void kernel_launch(void* const* d_in, const int* in_sizes, int n_in,
                              void* d_out, int out_size, void* d_ws, size_t ws_size,
                              hipStream_t stream) {
    (void)in_sizes; (void)n_in; (void)out_size; (void)ws_size;

    const float* x       = (const float*)d_in[0];
    const int*   targets = (const int*)d_in[1];
    const int*   sub     = (const int*)d_in[2];

    float* out  = (float*)d_out;
    float* dist = out + 1;                      // dist follows scalar loss

    float* xn      = (float*)d_ws;              // 8192*512 floats (16 MB)
    float* sq      = xn + (size_t)NROWS * DIMS; // 8192 floats
    float* rowLoss = sq + NROWS;                // 8192 floats

    normalize_kernel<<<NROWS, 256, 0, stream>>>(x, xn, sq);

    dim3 gGemm(NROWS / BN, NROWS / BM);         // (128, 64)
    dist_gemm_kernel<<<gGemm, 256, 0, stream>>>(xn, sq, dist);

    row_reduce_kernel<<<NROWS, 256, 0, stream>>>(dist, targets, sub, rowLoss);

    final_loss_kernel<<<1, 256, 0, stream>>>(rowLoss, out);
}